// SparseAdam_65214783422591
// MI455X (gfx1250) — compile-verified
//
#include <hip/hip_runtime.h>
#include <hip/hip_bf16.h>

#define LRc   1.0e-3f
#define B1c   0.9f
#define B2c   0.999f
#define EPSc  1.0e-8f
#define DD    128

#define LDS_AS __attribute__((address_space(3)))

typedef int v4i __attribute__((ext_vector_type(4)));

// gfx1250 async global->LDS path (ASYNCcnt), guarded so compilation never breaks.
#if defined(__HIP_DEVICE_COMPILE__) && defined(__gfx1250__) && \
    __has_builtin(__builtin_amdgcn_global_load_async_to_lds_b128)
#define USE_ASYNC_LDS 1
#else
#define USE_ASYNC_LDS 0
#endif

// ---------------- zeroing kernels (init cnt + gsum accumulators in d_out) ---
__global__ __launch_bounds__(256) void k_zero4(float4* __restrict__ p, long n4) {
    long i = (long)blockIdx.x * 256 + threadIdx.x;
    if (i < n4) p[i] = make_float4(0.f, 0.f, 0.f, 0.f);
}
__global__ __launch_bounds__(256) void k_zero1(float* __restrict__ p, long n) {
    long i = (long)blockIdx.x * 256 + threadIdx.x;
    if (i < n) p[i] = 0.f;
}

// ---------------- phase 1: segment-sum scatter via L2 f32 atomics -----------
// one wave32 per grad row; each lane: float4 load + 4 no-return atomic adds
__global__ __launch_bounds__(256) void k_scatter(const int* __restrict__ idx,
                                                 const float* __restrict__ grad,
                                                 float* gsum,   // aliases out_power
                                                 float* cnt,    // aliases out_step
                                                 long N) {
    long tid  = (long)blockIdx.x * 256 + threadIdx.x;
    long row  = tid >> 5;
    int  lane = (int)(threadIdx.x & 31u);
    if (row >= N) return;
    int v = idx[row];
    float4 g = *(const float4*)(grad + row * DD + lane * 4);
    float* dst = gsum + (size_t)v * DD + lane * 4;
    atomicAdd(dst + 0, g.x);
    atomicAdd(dst + 1, g.y);
    atomicAdd(dst + 2, g.z);
    atomicAdd(dst + 3, g.w);
    if (lane == 0) atomicAdd(cnt + v, 1.0f);
}

// ---------------- phase 2: per-vocab-row Adam update ------------------------
// one wave32 per vocab row; gsum row staged into LDS via async-to-LDS B128
__global__ __launch_bounds__(256) void k_update(const float* __restrict__ emb,
                                                const float* __restrict__ step_in,
                                                const float* __restrict__ mem_in,
                                                const float* __restrict__ pow_in,
                                                float* __restrict__ out_emb,
                                                float* step_cnt,   // cnt in, new_step out
                                                float* __restrict__ out_mem,
                                                float* pow_gsum,   // gsum in, new_power out
                                                long V) {
    long     tid  = (long)blockIdx.x * 256 + threadIdx.x;
    long     row  = tid >> 5;
    unsigned lane = threadIdx.x & 31u;
    unsigned wave = threadIdx.x >> 5;
    __shared__ float s_g[8 * DD];   // 4 KB: one gsum row per wave
    if (row >= V) return;

    const size_t off = (size_t)row * DD + lane * 4u;

    float4 g4;
#if USE_ASYNC_LDS
    {
        const float* src  = pow_gsum + off;
        float*       ldst = &s_g[wave * DD + lane * 4u];
        __builtin_amdgcn_global_load_async_to_lds_b128(
            (v4i*)src, (LDS_AS v4i*)ldst, 0, 0);
#if __has_builtin(__builtin_amdgcn_s_wait_asynccnt)
        __builtin_amdgcn_s_wait_asynccnt(0);
#else
        asm volatile("s_wait_asynccnt 0x0" ::: "memory");
#endif
        g4 = *(const float4*)ldst;   // ds_load_b128, wave-private region
    }
#else
    (void)s_g; (void)wave;
    g4 = *(const float4*)(pow_gsum + off);
#endif

    const float c       = step_cnt[row];          // accumulated count (float)
    const bool  touched = c > 0.0f;
    const float inv     = 1.0f / fmaxf(c, 1.0f);

    float st = step_in[row];
    float4 m = *(const float4*)(mem_in + off);
    float4 p = *(const float4*)(pow_in + off);
    float4 e = *(const float4*)(emb   + off);
    float4 nm = m, np = p, ne = e;

    if (touched) {                                // uniform per wave
        st += 1.0f;
        const float d1  = 1.0f - powf(B1c, st);
        const float d2  = 1.0f - powf(B2c, st);
        const float rd1 = LRc / d1;
#define UPD(comp)                                                         \
        {                                                                 \
            float ga  = g4.comp * inv;                                    \
            float mm  = B1c * m.comp + (1.0f - B1c) * ga;                 \
            float pp  = B2c * p.comp + (1.0f - B2c) * ga * ga;            \
            float stp = (rd1 * mm) / (sqrtf(pp / d2) + EPSc);             \
            nm.comp = mm; np.comp = pp; ne.comp = e.comp - stp;           \
        }
        UPD(x) UPD(y) UPD(z) UPD(w)
#undef UPD
    }

    *(float4*)(out_emb  + off) = ne;
    *(float4*)(out_mem  + off) = nm;
    *(float4*)(pow_gsum + off) = np;              // overwrite gsum with new_power
    if (lane == 0) step_cnt[row] = st;            // overwrite cnt with new_step
}

// ---------------------------------------------------------------------------
extern "C" void kernel_launch(void* const* d_in, const int* in_sizes, int n_in,
                              void* d_out, int out_size, void* d_ws, size_t ws_size,
                              hipStream_t stream) {
    const int*   idx  = (const int*)d_in[0];
    const float* grad = (const float*)d_in[1];
    const float* emb  = (const float*)d_in[2];
    const float* step = (const float*)d_in[3];
    const float* mem  = (const float*)d_in[4];
    const float* pw   = (const float*)d_in[5];

    const long N  = in_sizes[0];
    const long V  = in_sizes[3];
    const long VD = V * (long)DD;

    float* out      = (float*)d_out;
    float* out_emb  = out;                 // V*D
    float* out_step = out + VD;            // V      (doubles as cnt accumulator)
    float* out_mem  = out + VD + V;        // V*D
    float* out_pow  = out + VD + V + VD;   // V*D    (doubles as gsum accumulator)

    // phase 0: zero accumulator regions of d_out
    {
        long m4 = VD / 4;
        k_zero4<<<(unsigned)((m4 + 255) / 256), 256, 0, stream>>>((float4*)out_pow, m4);
        k_zero1<<<(unsigned)((V + 255) / 256), 256, 0, stream>>>(out_step, V);
    }
    // phase 1: segment sum (gsum, cnt) via atomics
    {
        long threads = N * 32;
        k_scatter<<<(unsigned)((threads + 255) / 256), 256, 0, stream>>>(
            idx, grad, out_pow, out_step, N);
    }
    // phase 2: per-row Adam update, writes all four outputs
    {
        long threads = V * 32;
        k_update<<<(unsigned)((threads + 255) / 256), 256, 0, stream>>>(
            emb, step, mem, pw, out_emb, out_step, out_mem, out_pow, V);
    }
}